// DatabaseFeatureExtractor_1185410974258
// MI455X (gfx1250) — compile-verified
//
#include <hip/hip_runtime.h>

#define BATCH 64
#define NPTS  500
#define DIM   2048
#define KCL   10
#define ITERS 10
#define DCHUNK 512
#define EPSV  1e-6f

typedef __attribute__((ext_vector_type(2))) float v2f;
typedef __attribute__((ext_vector_type(8))) float v8f;

// ---------------------------------------------------------------------------
// init: centroids = features[:, :K, :]  (centroids live in d_out)
// ---------------------------------------------------------------------------
__global__ __launch_bounds__(256) void kmeans_init(const float* __restrict__ feat,
                                                   float* __restrict__ cent) {
    int i = blockIdx.x * 256 + threadIdx.x;
    if (i >= BATCH * KCL * DIM) return;
    int d = i % DIM;
    int r = i / DIM;
    int k = r % KCL;
    int b = r / KCL;
    cent[i] = feat[((size_t)b * NPTS + k) * DIM + d];
}

// ---------------------------------------------------------------------------
// assign: per wave, a 16-point x 16-cluster tile of dot products via
// V_WMMA_F32_16X16X4_F32, contracting D=2048 in 512 steps of 4.
// argmin_k (|c_k|^2 - 2*<f,c_k>) == argmin_k |f - c_k|^2.
//
// Padded cluster columns (k=10..15) load a *duplicate* centroid row with no
// masking: their WMMA outputs are garbage-but-finite and are overridden with
// +inf before the argmin reduction, so no exec-mask manipulation is needed in
// the inner loop (keeps EXEC all-1s, as WMMA requires, and keeps the loop
// body at loads + wmma + 2 FMAs).
// 4 blocks/batch * 8 waves/block * 16 points/wave = 512 >= 500 points.
// ---------------------------------------------------------------------------
__global__ __launch_bounds__(256) void kmeans_assign(const float* __restrict__ feat,
                                                     const float* __restrict__ cent,
                                                     int* __restrict__ assign) {
    const int b     = blockIdx.x >> 2;
    const int wave  = threadIdx.x >> 5;
    const int lane  = threadIdx.x & 31;
    const int tile  = (blockIdx.x & 3) * 8 + wave;   // 0..31
    const int pbase = tile * 16;
    const int col   = lane & 15;   // A: point row within tile; B/C: cluster index
    const int half  = lane >> 4;   // selects d+0/d+1 (half 0) vs d+2/d+3 (half 1)

    const int arow = (pbase + col < NPTS) ? (pbase + col) : (NPTS - 1);
    const float* fptr = feat + ((size_t)b * NPTS + arow) * DIM + half * 2;
    const int brow = (col < KCL) ? col : (KCL - 1);
    const float* cptr = cent + ((size_t)b * KCL + brow) * DIM + half * 2;
    const bool kvalid = (col < KCL);

    v8f  acc = {};
    float csq = 0.0f;
#pragma unroll 8
    for (int d = 0; d < DIM; d += 4) {
        v2f a  = *(const v2f*)(fptr + d);
        v2f bb = *(const v2f*)(cptr + d);
        csq += bb.x * bb.x + bb.y * bb.y;
        acc = __builtin_amdgcn_wmma_f32_16x16x4_f32(false, a, false, bb,
                                                    (short)0, acc, false, false);
    }
    // combine the two half-wave partial |c_k|^2 sums (d parity 0/1 vs 2/3)
    csq += __shfl_xor(csq, 16, 32);

    // acc[j] = <f_{pbase + j + 8*half}, c_col>.  Reduce over the 16 lanes of
    // each half-wave (masks 8,4,2,1 never cross the half boundary).
#pragma unroll
    for (int j = 0; j < 8; ++j) {
        float score = kvalid ? (csq - 2.0f * acc[j]) : 3.4e38f;
        int   kidx  = col;
#pragma unroll
        for (int m = 8; m >= 1; m >>= 1) {
            float os = __shfl_xor(score, m, 32);
            int   ok = __shfl_xor(kidx,  m, 32);
            if (os < score || (os == score && ok < kidx)) { score = os; kidx = ok; }
        }
        const int point = pbase + j + 8 * half;
        if (col == 0 && point < NPTS) assign[b * NPTS + point] = kidx;
    }
}

// ---------------------------------------------------------------------------
// update: new centroid = segmented sum / max(count,1).  One block per
// (batch, 512-wide D chunk); LDS accumulator 10x512 (20 KB), each thread
// owns two D columns so no accumulation races.
// ---------------------------------------------------------------------------
__global__ __launch_bounds__(256) void kmeans_update(const float* __restrict__ feat,
                                                     const int* __restrict__ assign,
                                                     float* __restrict__ cent) {
    const int b     = blockIdx.x >> 2;
    const int dbase = (blockIdx.x & 3) * DCHUNK;
    const int tid   = threadIdx.x;

    __shared__ float s_acc[KCL * DCHUNK];
    __shared__ int   s_assign[NPTS];
    __shared__ int   s_cnt[KCL];

    for (int i = tid; i < KCL * DCHUNK; i += 256) s_acc[i] = 0.0f;
    if (tid < KCL) s_cnt[tid] = 0;
    for (int n = tid; n < NPTS; n += 256) s_assign[n] = assign[b * NPTS + n];
    __syncthreads();
    for (int n = tid; n < NPTS; n += 256) atomicAdd(&s_cnt[s_assign[n]], 1);
    __syncthreads();

    const float* fB = feat + (size_t)b * NPTS * DIM + dbase + tid * 2;
    for (int n = 0; n < NPTS; ++n) {
        const int k = s_assign[n];
        v2f v = *(const v2f*)(fB + (size_t)n * DIM);
        float* a = &s_acc[k * DCHUNK + tid * 2];
        a[0] += v.x;
        a[1] += v.y;
    }
    __syncthreads();

    for (int i = tid; i < KCL * DCHUNK; i += 256) {
        const int k = i / DCHUNK;
        const int d = i % DCHUNK;
        const int c = (s_cnt[k] > 1) ? s_cnt[k] : 1;
        cent[((size_t)b * KCL + k) * DIM + dbase + d] = s_acc[i] / (float)c;
    }
}

// ---------------------------------------------------------------------------
// gem: mean of clip(f,eps)^p per cluster, then ^(1/p); empty cluster falls
// back to the final centroid (in-place read-before-write of d_out).
// ---------------------------------------------------------------------------
__global__ __launch_bounds__(256) void kmeans_gem(const float* __restrict__ feat,
                                                  const int* __restrict__ assign,
                                                  const float* __restrict__ p_in,
                                                  float* __restrict__ out) {
    const int b     = blockIdx.x >> 2;
    const int dbase = (blockIdx.x & 3) * DCHUNK;
    const int tid   = threadIdx.x;
    const float p   = p_in[0];

    __shared__ float s_acc[KCL * DCHUNK];
    __shared__ int   s_assign[NPTS];
    __shared__ int   s_cnt[KCL];

    for (int i = tid; i < KCL * DCHUNK; i += 256) s_acc[i] = 0.0f;
    if (tid < KCL) s_cnt[tid] = 0;
    for (int n = tid; n < NPTS; n += 256) s_assign[n] = assign[b * NPTS + n];
    __syncthreads();
    for (int n = tid; n < NPTS; n += 256) atomicAdd(&s_cnt[s_assign[n]], 1);
    __syncthreads();

    const float* fB = feat + (size_t)b * NPTS * DIM + dbase + tid * 2;
    for (int n = 0; n < NPTS; ++n) {
        const int k = s_assign[n];
        v2f v = *(const v2f*)(fB + (size_t)n * DIM);
        float x0 = fmaxf(v.x, EPSV);
        float x1 = fmaxf(v.y, EPSV);
        float* a = &s_acc[k * DCHUNK + tid * 2];
        a[0] += __expf(p * __logf(x0));
        a[1] += __expf(p * __logf(x1));
    }
    __syncthreads();

    for (int i = tid; i < KCL * DCHUNK; i += 256) {
        const int k = i / DCHUNK;
        const int d = i % DCHUNK;
        const size_t o = ((size_t)b * KCL + k) * DIM + dbase + d;
        const int c = s_cnt[k];
        float res;
        if (c > 0) {
            float mean = s_acc[i] / (float)c;
            res = __expf(__logf(mean) / p);
        } else {
            res = out[o];   // final centroid fallback (== 0 for empty clusters)
        }
        out[o] = res;
    }
}

// ---------------------------------------------------------------------------
extern "C" void kernel_launch(void* const* d_in, const int* in_sizes, int n_in,
                              void* d_out, int out_size, void* d_ws, size_t ws_size,
                              hipStream_t stream) {
    const float* feat = (const float*)d_in[0];
    const float* p    = (const float*)d_in[1];
    float* cent_out   = (float*)d_out;       // B*K*D: centroids, then final output
    int*   assign     = (int*)d_ws;          // B*N ints

    kmeans_init<<<(BATCH * KCL * DIM + 255) / 256, 256, 0, stream>>>(feat, cent_out);

    for (int it = 0; it < ITERS; ++it) {
        kmeans_assign<<<BATCH * 4, 256, 0, stream>>>(feat, cent_out, assign);
        kmeans_update<<<BATCH * 4, 256, 0, stream>>>(feat, assign, cent_out);
    }

    kmeans_gem<<<BATCH * 4, 256, 0, stream>>>(feat, assign, p, cent_out);
}